// Gcn_Net_17858474016867
// MI455X (gfx1250) — compile-verified
//
#include <hip/hip_runtime.h>

#define N_NODES 100000
#define N_EDGES 3200000
#define F_IN    1433
#define F_HID   16
#define F_OUT   7

typedef __attribute__((ext_vector_type(2))) float v2f;
typedef __attribute__((ext_vector_type(8))) float v8f;

// ---------------------------------------------------------------- utilities
__global__ void zero_kernel(float* __restrict__ p, long n) {
    long i = (long)blockIdx.x * blockDim.x + threadIdx.x;
    long stride = (long)gridDim.x * blockDim.x;
    for (; i < n; i += stride) p[i] = 0.0f;
}

__global__ void degree_kernel(const int* __restrict__ src, const int* __restrict__ dst,
                              float* __restrict__ deg_out, float* __restrict__ deg_in) {
    int i = blockIdx.x * blockDim.x + threadIdx.x;
    if (i < N_EDGES) {
        atomicAdd(&deg_out[src[i]], 1.0f);
        atomicAdd(&deg_in[dst[i]], 1.0f);
    }
}

__global__ void norm_kernel(const float* __restrict__ deg_out, const float* __restrict__ deg_in,
                            float* __restrict__ ns, float* __restrict__ nd) {
    int i = blockIdx.x * blockDim.x + threadIdx.x;
    if (i < N_NODES) {
        ns[i] = rsqrtf(fmaxf(deg_out[i], 1.0f));
        nd[i] = rsqrtf(fmaxf(deg_in[i], 1.0f));
    }
}

// ---------------------------------------------------------------- layer 1 GEMM
// h1[16-row tile, 16] = (features * ns) @ W1 via V_WMMA_F32_16X16X4_F32.
// A layout (16x4 f32): lanes 0-15 hold K={k,k+1} of row M=lane; lanes 16-31 hold K={k+2,k+3}.
// B layout (4x16 f32): lane half selects K rows the same way; N = lane & 15.
// C/D layout: VGPR j -> M = j + 8*(lane>=16), N = lane & 15.
__global__ void gemm1_kernel(const float* __restrict__ feat, const float* __restrict__ W1,
                             const float* __restrict__ ns, float* __restrict__ h1) {
    const int lane   = threadIdx.x & 31;
    const int wave   = (blockIdx.x * blockDim.x + threadIdx.x) >> 5;
    const int nwaves = (gridDim.x * blockDim.x) >> 5;
    const int ntiles = N_NODES / 16;           // 6250, exact
    const int m      = lane & 15;
    const int koff   = (lane >> 4) << 1;       // 0 or 2

    for (int tile = wave; tile < ntiles; tile += nwaves) {
        const int row = tile * 16 + m;
        const float scale = ns[row];
        const float* __restrict__ arow = feat + (long)row * F_IN;

        v8f c = {};
        int k = 0;
        for (; k + 4 <= F_IN - 1; k += 4) {    // 358 full K=4 steps (K 0..1431)
            __builtin_prefetch(arow + k + 64, 0, 1);   // global_prefetch_b8 down the K walk
            v2f a, b;
            a.x = arow[k + koff]     * scale;
            a.y = arow[k + koff + 1] * scale;
            b.x = W1[(k + koff)     * F_HID + m];
            b.y = W1[(k + koff + 1) * F_HID + m];
            c = __builtin_amdgcn_wmma_f32_16x16x4_f32(false, a, false, b,
                                                      (short)0, c, false, false);
        }
        {   // tail: K = 1432, only lanes 0-15 (koff==0) carry data, rest contribute 0
            v2f a = {}, b = {};
            if (koff == 0) {
                a.x = arow[F_IN - 1] * scale;
                b.x = W1[(F_IN - 1) * F_HID + m];
            }
            c = __builtin_amdgcn_wmma_f32_16x16x4_f32(false, a, false, b,
                                                      (short)0, c, false, false);
        }

        const int crow = tile * 16 + ((lane >> 4) << 3);
        float* __restrict__ hrow = h1 + (long)crow * F_HID + m;
#pragma unroll
        for (int j = 0; j < 8; ++j) hrow[(long)j * F_HID] = c[j];
    }
}

// ---------------------------------------------------------------- edge scatter, layer 1
// 16 lanes per edge: agg1[dst][f] += h1[src][f]
__global__ void scatter1_kernel(const int* __restrict__ src, const int* __restrict__ dst,
                                const float* __restrict__ h1, float* __restrict__ agg1) {
    long tid = (long)blockIdx.x * blockDim.x + threadIdx.x;
    if (tid < (long)N_EDGES * F_HID) {
        int e = (int)(tid >> 4);
        int f = (int)(tid & 15);
        atomicAdd(&agg1[(long)dst[e] * F_HID + f], h1[(long)src[e] * F_HID + f]);
    }
}

// ---------------------------------------------------------------- layer 2 GEMM (fused epilogue of layer 1)
// A[row][k] = relu(agg1[row][k]*nd[row] + b1[k]) * ns[row];  B = W2 padded N 7->16.
__global__ void gemm2_kernel(const float* __restrict__ agg1, const float* __restrict__ W2,
                             const float* __restrict__ b1, const float* __restrict__ ns,
                             const float* __restrict__ nd, float* __restrict__ h2) {
    const int lane   = threadIdx.x & 31;
    const int wave   = (blockIdx.x * blockDim.x + threadIdx.x) >> 5;
    const int nwaves = (gridDim.x * blockDim.x) >> 5;
    const int ntiles = N_NODES / 16;
    const int m      = lane & 15;
    const int koff   = (lane >> 4) << 1;

    for (int tile = wave; tile < ntiles; tile += nwaves) {
        const int row = tile * 16 + m;
        const float sn = ns[row];
        const float dn = nd[row];
        const float* __restrict__ arow = agg1 + (long)row * F_HID;

        v8f c = {};
#pragma unroll
        for (int k = 0; k < F_HID; k += 4) {   // K = 16 -> 4 WMMA steps
            const int k0 = k + koff, k1 = k + koff + 1;
            v2f a, b;
            a.x = fmaxf(arow[k0] * dn + b1[k0], 0.0f) * sn;
            a.y = fmaxf(arow[k1] * dn + b1[k1], 0.0f) * sn;
            b.x = (m < F_OUT) ? W2[k0 * F_OUT + m] : 0.0f;
            b.y = (m < F_OUT) ? W2[k1 * F_OUT + m] : 0.0f;
            c = __builtin_amdgcn_wmma_f32_16x16x4_f32(false, a, false, b,
                                                      (short)0, c, false, false);
        }

        if (m < F_OUT) {
            const int crow = tile * 16 + ((lane >> 4) << 3);
            float* __restrict__ hrow = h2 + (long)crow * F_OUT + m;
#pragma unroll
            for (int j = 0; j < 8; ++j) hrow[(long)j * F_OUT] = c[j];
        }
    }
}

// ---------------------------------------------------------------- edge scatter, layer 2
// 8 lanes per edge, 7 active: agg2[dst][f] += h2[src][f]
__global__ void scatter2_kernel(const int* __restrict__ src, const int* __restrict__ dst,
                                const float* __restrict__ h2, float* __restrict__ agg2) {
    long tid = (long)blockIdx.x * blockDim.x + threadIdx.x;
    if (tid < (long)N_EDGES * 8) {
        int e = (int)(tid >> 3);
        int f = (int)(tid & 7);
        if (f < F_OUT)
            atomicAdd(&agg2[(long)dst[e] * F_OUT + f], h2[(long)src[e] * F_OUT + f]);
    }
}

// ---------------------------------------------------------------- final epilogue
__global__ void finish_kernel(const float* __restrict__ agg2, const float* __restrict__ nd,
                              const float* __restrict__ b2, float* __restrict__ out) {
    int i = blockIdx.x * blockDim.x + threadIdx.x;
    if (i < N_NODES * F_OUT) {
        int node = i / F_OUT;
        int f    = i - node * F_OUT;
        out[i] = agg2[i] * nd[node] + b2[f];
    }
}

// ---------------------------------------------------------------- launcher
extern "C" void kernel_launch(void* const* d_in, const int* in_sizes, int n_in,
                              void* d_out, int out_size, void* d_ws, size_t ws_size,
                              hipStream_t stream) {
    const float* feat = (const float*)d_in[0];
    const int*   src  = (const int*)  d_in[1];
    const int*   dst  = (const int*)  d_in[2];
    const float* W1   = (const float*)d_in[3];
    const float* b1   = (const float*)d_in[4];
    const float* W2   = (const float*)d_in[5];
    const float* b2   = (const float*)d_in[6];
    float* out = (float*)d_out;

    float* ws      = (float*)d_ws;
    float* deg_out = ws;
    float* deg_in  = ws +  1L * N_NODES;
    float* ns      = ws +  2L * N_NODES;
    float* nd      = ws +  3L * N_NODES;
    float* h1      = ws +  4L * N_NODES;   // [N,16]
    float* agg1    = ws + 20L * N_NODES;   // [N,16]
    float* h2      = ws + 36L * N_NODES;   // [N,7]
    float* agg2    = ws + 43L * N_NODES;   // [N,7]  (total 50*N floats = 20 MB)

    zero_kernel<<<2048, 256, 0, stream>>>(ws, 50L * N_NODES);

    degree_kernel<<<(N_EDGES + 255) / 256, 256, 0, stream>>>(src, dst, deg_out, deg_in);
    norm_kernel<<<(N_NODES + 255) / 256, 256, 0, stream>>>(deg_out, deg_in, ns, nd);

    const int gemm_blocks = (N_NODES / 16 + 7) / 8;   // 8 waves (tiles) per 256-thread block
    gemm1_kernel<<<gemm_blocks, 256, 0, stream>>>(feat, W1, ns, h1);

    scatter1_kernel<<<(int)(((long)N_EDGES * 16 + 255) / 256), 256, 0, stream>>>(src, dst, h1, agg1);

    gemm2_kernel<<<gemm_blocks, 256, 0, stream>>>(agg1, W2, b1, ns, nd, h2);

    scatter2_kernel<<<(int)(((long)N_EDGES * 8 + 255) / 256), 256, 0, stream>>>(src, dst, h2, agg2);

    finish_kernel<<<(N_NODES * F_OUT + 255) / 256, 256, 0, stream>>>(agg2, nd, b2, out);
}